// HeadUpdator_26448408609134
// MI455X (gfx1250) — compile-verified
//
#include <hip/hip_runtime.h>
#include <hip/hip_bf16.h>
#include <math.h>
#include <stdint.h>

#define BS 8
#define NC 80
#define CC 64
#define HH 256
#define WW 256
#define HP 128
#define WP 128
#define WCHUNK 64
#define ROWS_PER_WG 8
#define SPAD 68                 // LDS row stride (floats): 16B-aligned rows,
                                // bank = (4*row + k) % 64 -> conflict-free frags
#define FEATBUF (CC * SPAD)

typedef float v2f __attribute__((ext_vector_type(2)));
typedef float v8f __attribute__((ext_vector_type(8)));
typedef int   v4i __attribute__((ext_vector_type(4)));

#if defined(__has_builtin)
#if __has_builtin(__builtin_amdgcn_global_load_async_to_lds_b128)
#define HAVE_ASYNC_LDS 1
#endif
#endif
#ifndef HAVE_ASYNC_LDS
#define HAVE_ASYNC_LDS 0
#endif

#if HAVE_ASYNC_LDS
typedef __attribute__((address_space(1))) v4i gv4i;   // global v4i (AS1)
typedef __attribute__((address_space(3))) v4i lv4i;   // LDS v4i (AS3)
#endif

__device__ inline void wait_async_zero() {
#if HAVE_ASYNC_LDS
#if __has_builtin(__builtin_amdgcn_s_wait_asynccnt)
    __builtin_amdgcn_s_wait_asynccnt(0);
#else
    asm volatile("s_wait_asynccnt 0x0" ::: "memory");
#endif
#endif
}

// ---------------------------------------------------------------------------
// Kernel 0: zero the assemble workspace (must be re-zeroed every call).
// ---------------------------------------------------------------------------
__global__ void zero_ws_kernel(float* __restrict__ p, int n) {
    int i = blockIdx.x * blockDim.x + threadIdx.x;
    if (i < n) p[i] = 0.0f;
}

// ---------------------------------------------------------------------------
// Stage one feat chunk (64 channels x 64 pixels) into LDS.
// Async ASYNCcnt-tracked B128 copies when available (bypasses VGPRs).
// ---------------------------------------------------------------------------
__device__ inline void load_feat_chunk(const float* __restrict__ feat,
                                       int b, int h, int w0,
                                       float* __restrict__ sbuf, int tid) {
    for (int idx = tid; idx < CC * WCHUNK / 4; idx += 640) {
        const int c  = idx >> 4;        // channel
        const int wq = idx & 15;        // 16B quad within row
        const float* src =
            &feat[(((size_t)b * CC + c) * HH + h) * WW + (w0 + wq * 4)];
        float* dst = &sbuf[c * SPAD + wq * 4];
#if HAVE_ASYNC_LDS
        __builtin_amdgcn_global_load_async_to_lds_b128(
            (gv4i*)(uintptr_t)src, (lv4i*)dst, 0, 0);
#else
        *(float4*)dst = *(const float4*)src;
#endif
    }
}

// ---------------------------------------------------------------------------
// Compute one upsample+sigmoid pred chunk (80 classes x 64 pixels) into LDS.
// Half-pixel-center bilinear with edge clamp (jax.image.resize semantics).
// ---------------------------------------------------------------------------
__device__ inline void compute_pred_chunk(const float* __restrict__ pred,
                                          int b, int h, int w0,
                                          float* __restrict__ spred, int tid) {
    const float fy = h * 0.5f - 0.25f;
    const float iy = floorf(fy);
    const float ty = fy - iy;
    int y0 = (int)iy;
    int y1 = y0 + 1;
    y0 = y0 < 0 ? 0 : y0;
    y1 = y1 > (HP - 1) ? (HP - 1) : y1;

    for (int idx = tid; idx < NC * WCHUNK; idx += 640) {
        const int n  = idx >> 6;
        const int wl = idx & 63;
        const float fx  = (float)(w0 + wl) * 0.5f - 0.25f;
        const float ixf = floorf(fx);
        const float tx  = fx - ixf;
        int x0 = (int)ixf;
        int x1 = x0 + 1;
        x0 = x0 < 0 ? 0 : x0;
        x1 = x1 > (WP - 1) ? (WP - 1) : x1;
        const float* pb = pred + ((size_t)b * NC + n) * (HP * WP);
        const float p00 = pb[y0 * WP + x0];
        const float p01 = pb[y0 * WP + x1];
        const float p10 = pb[y1 * WP + x0];
        const float p11 = pb[y1 * WP + x1];
        const float v = (1.f - ty) * ((1.f - tx) * p00 + tx * p01) +
                        ty         * ((1.f - tx) * p10 + tx * p11);
        spred[n * SPAD + wl] = 1.f / (1.f + __expf(-v));
    }
}

// ---------------------------------------------------------------------------
// Kernel A: fused bilinear-upsample + sigmoid + einsum('bnhw,bchw->bnc').
// 640 threads = 20 waves; wave (nt,ct) owns a 16x16 tile of the 80x64 result.
// Double-buffered feat tiles: async copy of chunk t+1 overlaps WMMA on chunk t.
// ---------------------------------------------------------------------------
__global__ __launch_bounds__(640) void assemble_kernel(
    const float* __restrict__ feat,   // (BS, CC, HH, WW)
    const float* __restrict__ pred,   // (BS, NC, HP, WP)
    float* __restrict__ assemble)     // (BS, NC, CC) accumulated via atomics
{
    __shared__ __align__(16) float sfeat[2 * FEATBUF]; // double-buffered [c][w]
    __shared__ __align__(16) float spred[NC * SPAD];   // [n][w]

    const int b    = blockIdx.y;
    const int rg   = blockIdx.x;               // row group (8 rows each)
    const int tid  = threadIdx.x;
    const int wave = tid >> 5;
    const int lane = tid & 31;
    const int nt   = wave >> 2;                // 0..4 : n-tile
    const int ct   = wave & 3;                 // 0..3 : c-tile
    const int lhalf = lane >> 4;               // 0 or 1
    const int l15   = lane & 15;

    // WMMA f32 16x16x4 operand layouts (ISA 7.12.2):
    //  A (16x4):  lanes 0-15 M=lane, lanes 16-31 M=lane-16; K = VGPR + 2*lhalf.
    //  B (4x16):  lanes 0-15 N=lane, lanes 16-31 N=lane-16; K = VGPR + 2*lhalf.
    const int arow = nt * 16 + l15;            // n row for A fragment
    const int brow = ct * 16 + l15;            // c column (feat row) for B
    const int koff = lhalf * 2;

    v8f acc = {0.f, 0.f, 0.f, 0.f, 0.f, 0.f, 0.f, 0.f};

    const int NCHUNK = ROWS_PER_WG * (WW / WCHUNK);  // 32 chunks per WG
    const int h0 = rg * ROWS_PER_WG;

    // Prologue: stage chunk 0.
    load_feat_chunk(feat, b, h0, 0, sfeat, tid);
    compute_pred_chunk(pred, b, h0, 0, spred, tid);
    wait_async_zero();
    __syncthreads();

    for (int t = 0; t < NCHUNK; ++t) {
        const int cur = t & 1;

        // Kick off the async feat copy for chunk t+1 into the other buffer
        // (free since the end-of-iteration barrier of t-1). Overlaps the WMMAs.
        if (t + 1 < NCHUNK) {
            const int hn  = h0 + ((t + 1) >> 2);
            const int w0n = ((t + 1) & 3) * WCHUNK;
            load_feat_chunk(feat, b, hn, w0n, sfeat + (cur ^ 1) * FEATBUF, tid);
        }

        // GEMM phase: reduce 64 pixels in steps of K=4.
        const float* fbuf = sfeat + cur * FEATBUF;
        #pragma unroll
        for (int k = 0; k < WCHUNK; k += 4) {
            const int lk = k + koff;
            v2f A, B;
            A.x = spred[arow * SPAD + lk];
            A.y = spred[arow * SPAD + lk + 1];
            B.x = fbuf[brow * SPAD + lk];
            B.y = fbuf[brow * SPAD + lk + 1];
            acc = __builtin_amdgcn_wmma_f32_16x16x4_f32(
                      false, A, false, B, (short)0, acc, false, false);
        }
        __syncthreads();   // all waves done reading spred & sfeat[cur]

        // Recompute the (single-buffered) pred chunk for t+1 while the async
        // feat copy drains.
        if (t + 1 < NCHUNK) {
            const int hn  = h0 + ((t + 1) >> 2);
            const int w0n = ((t + 1) & 3) * WCHUNK;
            compute_pred_chunk(pred, b, hn, w0n, spred, tid);
        }
        wait_async_zero();
        __syncthreads();
    }

    // D layout (16x16 f32): VGPR r -> M = r (+8 for lanes 16-31), N = lane&15.
    const int nbase = nt * 16 + lhalf * 8;
    const int cidx  = ct * 16 + l15;
    #pragma unroll
    for (int r = 0; r < 8; ++r) {
        atomicAdd(&assemble[((size_t)b * NC + (nbase + r)) * CC + cidx], acc[r]);
    }
}

// ---------------------------------------------------------------------------
// Kernel B: per-row gating network. One 128-thread block per (b, n) row.
// ---------------------------------------------------------------------------
__device__ inline float ln_apply(const float* __restrict__ x, int j,
                                 const float* __restrict__ g,
                                 const float* __restrict__ be) {
    float m = 0.f;
    for (int k = 0; k < CC; ++k) m += x[k];
    m *= (1.f / CC);
    float v = 0.f;
    for (int k = 0; k < CC; ++k) { const float d = x[k] - m; v += d * d; }
    v *= (1.f / CC);
    return (x[j] - m) * rsqrtf(v + 1e-5f) * g[j] + be[j];
}

__global__ __launch_bounds__(128) void head_update_kernel(
    const float* __restrict__ assemble, const float* __restrict__ head,
    const float* __restrict__ W_pred_t, const float* __restrict__ b_pred_t,
    const float* __restrict__ W_head_t, const float* __restrict__ b_head_t,
    const float* __restrict__ W_pred_g, const float* __restrict__ b_pred_g,
    const float* __restrict__ W_head_g, const float* __restrict__ b_head_g,
    const float* __restrict__ W_fc,     const float* __restrict__ b_fc,
    const float* __restrict__ g_pin,  const float* __restrict__ be_pin,
    const float* __restrict__ g_hin,  const float* __restrict__ be_hin,
    const float* __restrict__ g_pout, const float* __restrict__ be_pout,
    const float* __restrict__ g_hout, const float* __restrict__ be_hout,
    const float* __restrict__ g_fc,   const float* __restrict__ be_fc,
    float* __restrict__ out)
{
    __shared__ float a_s[CC], hd_s[CC];
    __shared__ float pf_s[2 * CC], hf_s[2 * CC];
    __shared__ float gate_s[CC], gh_s[CC], gp_s[CC];
    __shared__ float upd_s[CC], fc_s[CC];

    const int i = blockIdx.x;        // row in [0, BS*NC)
    const int t = threadIdx.x;       // 0..127

    if (t < CC) {
        a_s[t]  = assemble[(size_t)i * CC + t];
        hd_s[t] = head[(size_t)i * CC + t];      // head[b,n,c,0]
    }
    __syncthreads();

    // pf = a @ W_pred_t + b_pred_t ; hf = hd @ W_head_t + b_head_t  (128 wide)
    {
        float accp = b_pred_t[t];
        float acch = b_head_t[t];
        for (int k = 0; k < CC; ++k) {
            accp += a_s[k]  * W_pred_t[k * (2 * CC) + t];
            acch += hd_s[k] * W_head_t[k * (2 * CC) + t];
        }
        pf_s[t] = accp;
        hf_s[t] = acch;
    }
    __syncthreads();

    if (t < CC) gate_s[t] = hf_s[t] * pf_s[t];   // hf_in * pf_in
    __syncthreads();

    if (t < CC) {
        float gh = b_head_g[t];
        float gp = b_pred_g[t];
        for (int k = 0; k < CC; ++k) {
            gh += gate_s[k] * W_head_g[k * CC + t];
            gp += gate_s[k] * W_pred_g[k * CC + t];
        }
        gh_s[t] = gh;
        gp_s[t] = gp;
    }
    __syncthreads();

    float head_gate = 0.f, pred_gate = 0.f, hfo = 0.f, pfo = 0.f;
    if (t < CC) {
        head_gate = 1.f / (1.f + __expf(-ln_apply(gh_s, t, g_hin, be_hin)));
        pred_gate = 1.f / (1.f + __expf(-ln_apply(gp_s, t, g_pin, be_pin)));
        hfo = ln_apply(hf_s + CC, t, g_hout, be_hout);
        pfo = ln_apply(pf_s + CC, t, g_pout, be_pout);
    }
    __syncthreads();

    if (t < CC) upd_s[t] = pred_gate * pfo + head_gate * hfo;
    __syncthreads();

    if (t < CC) {
        float f = b_fc[t];
        for (int k = 0; k < CC; ++k) f += upd_s[k] * W_fc[k * CC + t];
        fc_s[t] = f;
    }
    __syncthreads();

    if (t < CC) {
        const float y = ln_apply(fc_s, t, g_fc, be_fc);
        out[(size_t)i * CC + t] = y > 0.f ? y : 0.f;   // ReLU
    }
}

// ---------------------------------------------------------------------------
extern "C" void kernel_launch(void* const* d_in, const int* in_sizes, int n_in,
                              void* d_out, int out_size, void* d_ws, size_t ws_size,
                              hipStream_t stream) {
    const float* feat     = (const float*)d_in[0];
    const float* head     = (const float*)d_in[1];
    const float* pred     = (const float*)d_in[2];
    const float* W_pred_t = (const float*)d_in[3];
    const float* b_pred_t = (const float*)d_in[4];
    const float* W_head_t = (const float*)d_in[5];
    const float* b_head_t = (const float*)d_in[6];
    const float* W_pred_g = (const float*)d_in[7];
    const float* b_pred_g = (const float*)d_in[8];
    const float* W_head_g = (const float*)d_in[9];
    const float* b_head_g = (const float*)d_in[10];
    const float* W_fc     = (const float*)d_in[11];
    const float* b_fc     = (const float*)d_in[12];
    const float* g_pin    = (const float*)d_in[13];
    const float* be_pin   = (const float*)d_in[14];
    const float* g_hin    = (const float*)d_in[15];
    const float* be_hin   = (const float*)d_in[16];
    const float* g_pout   = (const float*)d_in[17];
    const float* be_pout  = (const float*)d_in[18];
    const float* g_hout   = (const float*)d_in[19];
    const float* be_hout  = (const float*)d_in[20];
    const float* g_fc     = (const float*)d_in[21];
    const float* be_fc    = (const float*)d_in[22];

    float* out      = (float*)d_out;
    float* assemble = (float*)d_ws;          // BS*NC*CC floats = 160 KiB
    const int an = BS * NC * CC;

    zero_ws_kernel<<<(an + 255) / 256, 256, 0, stream>>>(assemble, an);

    dim3 gridA(HH / ROWS_PER_WG, BS);        // (32, 8)
    assemble_kernel<<<gridA, 640, 0, stream>>>(feat, pred, assemble);

    head_update_kernel<<<BS * NC, 128, 0, stream>>>(
        assemble, head,
        W_pred_t, b_pred_t, W_head_t, b_head_t,
        W_pred_g, b_pred_g, W_head_g, b_head_g,
        W_fc, b_fc,
        g_pin, be_pin, g_hin, be_hin,
        g_pout, be_pout, g_hout, be_hout,
        g_fc, be_fc,
        out);
}